// QuantumFeedForward_65481071397683
// MI455X (gfx1250) — compile-verified
//
#include <hip/hip_runtime.h>
#include <hip/hip_bf16.h>

#define NQ   10
#define NL   4
#define FFN  4096
#define EDIM 1024
#define BT   (8 * 2048)          // B*T rows
#define NSTATE (1 << NQ)         // 1024 amplitudes

typedef float v2f __attribute__((ext_vector_type(2)));
typedef float v8f __attribute__((ext_vector_type(8)));

// ---------------------------------------------------------------------------
// Kernel 1: 10-qubit statevector simulation -> z[NQ] expectation values.
// One block, 512 threads (16 wave32s). State in LDS (8 KB).
// Qubit q <-> flat-index bit (NQ-1-q)  (axis 0 of the (2,)*n tensor is MSB).
// ---------------------------------------------------------------------------
__global__ void qffn_circuit_kernel(const float* __restrict__ params,
                                    float* __restrict__ z_out) {
    __shared__ float re[NSTATE];
    __shared__ float im[NSTATE];
    const int tid = threadIdx.x;            // 0..511

    for (int i = tid; i < NSTATE; i += 512) { re[i] = 0.0f; im[i] = 0.0f; }
    if (tid == 0) re[0] = 1.0f;
    __syncthreads();

    for (int l = 0; l < NL; ++l) {
        // --- single-qubit RX, RY, RZ on each qubit (fused per pair) ---
        for (int q = 0; q < NQ; ++q) {
            const int b   = NQ - 1 - q;
            const float thx = params[(l * NQ + q) * 3 + 0];
            const float thy = params[(l * NQ + q) * 3 + 1];
            const float thz = params[(l * NQ + q) * 3 + 2];

            const int low = tid & ((1 << b) - 1);
            const int i0  = ((tid >> b) << (b + 1)) | low;
            const int i1  = i0 | (1 << b);

            float r0 = re[i0], m0 = im[i0];
            float r1 = re[i1], m1 = im[i1];

            // RX: [[c,-is],[-is,c]]
            {
                float c = cosf(0.5f * thx), s = sinf(0.5f * thx);
                float nr0 = c * r0 + s * m1, nm0 = c * m0 - s * r1;
                float nr1 = c * r1 + s * m0, nm1 = c * m1 - s * r0;
                r0 = nr0; m0 = nm0; r1 = nr1; m1 = nm1;
            }
            // RY: [[c,-s],[s,c]] (real)
            {
                float c = cosf(0.5f * thy), s = sinf(0.5f * thy);
                float nr0 = c * r0 - s * r1, nm0 = c * m0 - s * m1;
                float nr1 = s * r0 + c * r1, nm1 = s * m0 + c * m1;
                r0 = nr0; m0 = nm0; r1 = nr1; m1 = nm1;
            }
            // RZ: diag(e^{-i th/2}, e^{+i th/2})
            {
                float c = cosf(0.5f * thz), s = sinf(0.5f * thz);
                float nr0 = r0 * c + m0 * s, nm0 = m0 * c - r0 * s;
                float nr1 = r1 * c - m1 * s, nm1 = m1 * c + r1 * s;
                r0 = nr0; m0 = nm0; r1 = nr1; m1 = nm1;
            }
            re[i0] = r0; im[i0] = m0;
            re[i1] = r1; im[i1] = m1;
            __syncthreads();
        }
        // --- CNOT ring: (0,1),(1,2),...,(8,9),(9,0) ---
        for (int g = 0; g < NQ; ++g) {
            const int cq = (g < NQ - 1) ? g : NQ - 1;
            const int tq = (g < NQ - 1) ? g + 1 : 0;
            const int bc = NQ - 1 - cq;
            const int bt = NQ - 1 - tq;
            if (tid < (NSTATE >> 2)) {       // 256 swap pairs
                const int blo = (bc < bt) ? bc : bt;
                const int bhi = (bc < bt) ? bt : bc;
                int x = tid;
                x = ((x >> blo) << (blo + 1)) | (x & ((1 << blo) - 1));
                x = ((x >> bhi) << (bhi + 1)) | (x & ((1 << bhi) - 1));
                const int i0 = x | (1 << bc);          // control=1, target=0
                const int i1 = i0 | (1 << bt);         // control=1, target=1
                float r = re[i0], m = im[i0];
                re[i0] = re[i1]; im[i0] = im[i1];
                re[i1] = r;      im[i1] = m;
            }
            __syncthreads();
        }
    }

    // <Z_q> = 1 - 2*p1 ; deterministic serial reduction (one thread per qubit)
    if (tid < NQ) {
        const int b = NQ - 1 - tid;
        float s = 0.0f;
        for (int j = 0; j < (NSTATE >> 1); ++j) {
            const int low = j & ((1 << b) - 1);
            const int idx = (((j >> b) << (b + 1)) | low) | (1 << b);
            s += re[idx] * re[idx] + im[idx] * im[idx];
        }
        z_out[tid] = 1.0f - 2.0f * s;
    }
}

// ---------------------------------------------------------------------------
// Kernel 2: h[f] = relu( sum_q W1[f,q]*z[q] + b1[f] ), f in [0,FFN)
// ---------------------------------------------------------------------------
__global__ void qffn_h_kernel(const float* __restrict__ W1,
                              const float* __restrict__ b1,
                              const float* __restrict__ z,
                              float* __restrict__ h) {
    const int f = blockIdx.x * blockDim.x + threadIdx.x;
    if (f < FFN) {
        float acc = b1[f];
        #pragma unroll
        for (int q = 0; q < NQ; ++q) acc += W1[f * NQ + q] * z[q];
        h[f] = fmaxf(acc, 0.0f);
    }
}

// ---------------------------------------------------------------------------
// Kernel 3: y[e] = sum_f W2[e,f]*h[f] + b2[e] via V_WMMA_F32_16X16X4_F32.
// Block = 128 threads (4 waves); each block owns 16 rows of W2; each wave
// covers a K-quarter (1024), partials combined through LDS deterministically.
// A-layout (16x4 f32): lanes 0-15 hold K=0,1 in v0,v1; lanes 16-31 hold K=2,3.
// B (4x16) has all columns equal to the h chunk -> all D columns equal y.
// D-layout: lane 0 = column N=0, M=0..7 in acc[0..7]; lane 16 = M=8..15.
// ---------------------------------------------------------------------------
__global__ void qffn_y_wmma_kernel(const float* __restrict__ W2,
                                   const float* __restrict__ b2,
                                   const float* __restrict__ h,
                                   float* __restrict__ y) {
    __shared__ float part[4][16];
    const int lane  = threadIdx.x & 31;
    const int wave  = threadIdx.x >> 5;
    const int r0    = blockIdx.x * 16;
    const int row   = r0 + (lane & 15);
    const int khalf = (lane >> 4) * 2;       // 0 for lanes 0-15, 2 for 16-31
    const int KW    = FFN / 4;               // K per wave
    const int kbase = wave * KW;

    v8f acc = {};
    for (int k = 0; k < KW; k += 4) {
        const int kk = kbase + k + khalf;
        const float* ap = W2 + (size_t)row * FFN + kk;
        v2f a, b;
        a.x = ap[0];  a.y = ap[1];
        b.x = h[kk];  b.y = h[kk + 1];
        acc = __builtin_amdgcn_wmma_f32_16x16x4_f32(
            /*neg_a=*/false, a, /*neg_b=*/false, b,
            /*c_mod=*/(short)0, acc, /*reuse_a=*/false, /*reuse_b=*/false);
    }

    if ((lane & 15) == 0) {                  // lanes 0 and 16: column N=0
        const int mbase = (lane < 16) ? 0 : 8;
        #pragma unroll
        for (int v = 0; v < 8; ++v) part[wave][mbase + v] = acc[v];
    }
    __syncthreads();
    if (threadIdx.x < 16) {
        const int m = threadIdx.x;
        y[r0 + m] = part[0][m] + part[1][m] + part[2][m] + part[3][m]
                  + b2[r0 + m];
    }
}

// ---------------------------------------------------------------------------
// Kernel 4: broadcast y_row (1024 f32 = 256 float4) to all 16384 rows.
// 256 blocks x 256 threads; each thread holds one float4 of y and streams
// 64 fully-coalesced row writes (dominant cost: 64 MB @ 23.3 TB/s ~ 2.7 us).
// ---------------------------------------------------------------------------
__global__ void qffn_broadcast_kernel(const float* __restrict__ y,
                                      float* __restrict__ out) {
    const int tid = threadIdx.x;             // 0..255  (one float4 per thread)
    const float4 v = ((const float4*)y)[tid];
    float4* o = (float4*)out + (size_t)blockIdx.x * 64u * 256u + tid;
    #pragma unroll
    for (int r = 0; r < 64; ++r) {
        o[(size_t)r * 256u] = v;
    }
}

// ---------------------------------------------------------------------------
// inputs (setup_inputs order): 0:x 1:W_in 2:b_in 3:params 4:W1 5:b1 6:W2 7:b2
// x / W_in / b_in are dead in the reference (the qnode ignores its x input).
// ---------------------------------------------------------------------------
extern "C" void kernel_launch(void* const* d_in, const int* in_sizes, int n_in,
                              void* d_out, int out_size, void* d_ws, size_t ws_size,
                              hipStream_t stream) {
    (void)in_sizes; (void)n_in; (void)out_size; (void)ws_size;
    const float* params = (const float*)d_in[3];
    const float* W1     = (const float*)d_in[4];
    const float* b1     = (const float*)d_in[5];
    const float* W2     = (const float*)d_in[6];
    const float* b2     = (const float*)d_in[7];

    float* ws = (float*)d_ws;
    float* z  = ws;                 // 10 floats
    float* h  = ws + 64;            // 4096 floats
    float* y  = ws + 64 + FFN;      // 1024 floats

    qffn_circuit_kernel  <<<1,   512, 0, stream>>>(params, z);
    qffn_h_kernel        <<<FFN / 256, 256, 0, stream>>>(W1, b1, z, h);
    qffn_y_wmma_kernel   <<<EDIM / 16, 128, 0, stream>>>(W2, b2, h, y);
    qffn_broadcast_kernel<<<BT / 64, 256, 0, stream>>>(y, (float*)d_out);
}